// NMP_49718541418978
// MI455X (gfx1250) — compile-verified
//
#include <hip/hip_runtime.h>
#include <math.h>

typedef __attribute__((ext_vector_type(2))) float v2f;
typedef __attribute__((ext_vector_type(8))) float v8f;

static constexpr int Nn  = 10000;   // nodes (exactly 625 * 16)
static constexpr int Ne  = 320000;  // edges
static constexpr int Hd  = 32;      // hidden width
static constexpr int Gg  = 64;      // graphs

// ---------------------------------------------------------------------------
// fp32 WMMA: D(16x16) = A(16x4) * B(4x16) + C   (V_WMMA_F32_16X16X4_F32)
// ---------------------------------------------------------------------------
__device__ __forceinline__ v8f wmma4(v2f a, v2f b, v8f c) {
  // (neg_a, A, neg_b, B, c_mod, C, reuse_a, reuse_b)
  return __builtin_amdgcn_wmma_f32_16x16x4_f32(false, a, false, b, (short)0, c,
                                               false, false);
}

// C0 = A@W0, C1 = A@W1, C2 = A@W2 ; A:[Nn x K] row-major, W:[K x 32] row-major.
// One wave handles one 16-row tile and all 32 output columns (2 col-tiles),
// reading A once for all three weight matrices.
template <int K>
__global__ __launch_bounds__(128) void gemm3_wmma(
    const float* __restrict__ A, const float* __restrict__ W0,
    const float* __restrict__ W1, const float* __restrict__ W2,
    float* __restrict__ C0, float* __restrict__ C1, float* __restrict__ C2) {
  const int lane = threadIdx.x & 31;
  const int wave = threadIdx.x >> 5;
  const int tile = blockIdx.x * 4 + wave;
  const int ntiles = Nn / 16;
  if (tile >= ntiles) return;          // wave-uniform: EXEC stays all-ones

  const int half = lane >> 4;          // 0: lanes 0-15, 1: lanes 16-31
  const int l15  = lane & 15;
  const int arow = tile * 16 + l15;

  v8f acc00 = {}; v8f acc01 = {};
  v8f acc10 = {}; v8f acc11 = {};
  v8f acc20 = {}; v8f acc21 = {};

#pragma unroll
  for (int k0 = 0; k0 < K; k0 += 4) {
    const int ka = k0 + 2 * half;      // A: VGPR0/1 hold K={ka, ka+1}
    v2f av;
    av.x = (ka     < K) ? A[arow * K + ka    ] : 0.0f;
    av.y = (ka + 1 < K) ? A[arow * K + ka + 1] : 0.0f;

    const int c0 = l15;                // B col-tile 0
    const int c1 = l15 + 16;           // B col-tile 1
    v2f b00, b01, b10, b11, b20, b21;
    b00.x = (ka     < K) ? W0[(ka    ) * Hd + c0] : 0.0f;
    b00.y = (ka + 1 < K) ? W0[(ka + 1) * Hd + c0] : 0.0f;
    b01.x = (ka     < K) ? W0[(ka    ) * Hd + c1] : 0.0f;
    b01.y = (ka + 1 < K) ? W0[(ka + 1) * Hd + c1] : 0.0f;
    b10.x = (ka     < K) ? W1[(ka    ) * Hd + c0] : 0.0f;
    b10.y = (ka + 1 < K) ? W1[(ka + 1) * Hd + c0] : 0.0f;
    b11.x = (ka     < K) ? W1[(ka    ) * Hd + c1] : 0.0f;
    b11.y = (ka + 1 < K) ? W1[(ka + 1) * Hd + c1] : 0.0f;
    b20.x = (ka     < K) ? W2[(ka    ) * Hd + c0] : 0.0f;
    b20.y = (ka + 1 < K) ? W2[(ka + 1) * Hd + c0] : 0.0f;
    b21.x = (ka     < K) ? W2[(ka    ) * Hd + c1] : 0.0f;
    b21.y = (ka + 1 < K) ? W2[(ka + 1) * Hd + c1] : 0.0f;

    acc00 = wmma4(av, b00, acc00);
    acc01 = wmma4(av, b01, acc01);
    acc10 = wmma4(av, b10, acc10);
    acc11 = wmma4(av, b11, acc11);
    acc20 = wmma4(av, b20, acc20);
    acc21 = wmma4(av, b21, acc21);
  }

  // C/D layout: VGPR r, lanes 0-15 -> row r, lanes 16-31 -> row r+8
  const int rbase = tile * 16 + 8 * half;
  const int ccol  = l15;
#pragma unroll
  for (int r = 0; r < 8; ++r) {
    C0[(rbase + r) * Hd + ccol     ] = acc00[r];
    C0[(rbase + r) * Hd + ccol + 16] = acc01[r];
    C1[(rbase + r) * Hd + ccol     ] = acc10[r];
    C1[(rbase + r) * Hd + ccol + 16] = acc11[r];
    C2[(rbase + r) * Hd + ccol     ] = acc20[r];
    C2[(rbase + r) * Hd + ccol + 16] = acc21[r];
  }
}

// ---------------------------------------------------------------------------
__global__ void fill_zero(float* __restrict__ p, int n) {
  int t = blockIdx.x * blockDim.x + threadIdx.x;
  if (t < n) p[t] = 0.0f;
}

// agg[dst] += a_e * P[src] + Q[src]; thread per (edge, 4-feature group)
__global__ void edge_scatter(const int* __restrict__ ei,
                             const float* __restrict__ ea,
                             const float* __restrict__ P,
                             const float* __restrict__ Q,
                             float* __restrict__ agg) {
  int tid = blockIdx.x * blockDim.x + threadIdx.x;
  if (tid >= Ne * 8) return;
  int e = tid >> 3;
  int f = (tid & 7) << 2;
  int s = ei[e];
  int d = ei[Ne + e];
  float a = ea[e];
  float4 p = *reinterpret_cast<const float4*>(P + s * Hd + f);
  float4 q = *reinterpret_cast<const float4*>(Q + s * Hd + f);
  float* o = agg + d * Hd + f;
  atomicAdd(o + 0, fmaf(a, p.x, q.x));
  atomicAdd(o + 1, fmaf(a, p.y, q.y));
  atomicAdd(o + 2, fmaf(a, p.z, q.z));
  atomicAdd(o + 3, fmaf(a, p.w, q.w));
}

// h = relu(agg + R + bias)
__global__ void node_update(const float* __restrict__ agg,
                            const float* __restrict__ R,
                            const float* __restrict__ bias,
                            float* __restrict__ hout) {
  int t = blockIdx.x * blockDim.x + threadIdx.x;
  if (t >= Nn * Hd) return;
  float v = agg[t] + R[t] + bias[t & (Hd - 1)];
  hout[t] = v > 0.0f ? v : 0.0f;
}

// add / count / max pooling (max via u32 bit pattern: valid for ReLU outputs)
__global__ void pool_kernel(const float* __restrict__ h,
                            const int* __restrict__ batch,
                            float* __restrict__ addp,
                            unsigned int* __restrict__ maxb,
                            float* __restrict__ cnt) {
  int t = blockIdx.x * blockDim.x + threadIdx.x;
  if (t >= Nn * Hd) return;
  int n = t >> 5, f = t & 31;
  int b = batch[n];
  float v = h[t];
  atomicAdd(&addp[b * Hd + f], v);
  atomicMax(&maxb[b * Hd + f], __float_as_uint(v));
  if (f == 0) atomicAdd(&cnt[b], 1.0f);
}

// readout: [add|mean|max] (96) -> relu Linear(96,32) -> Linear(32,2) -> logsoftmax
__global__ void readout(const float* __restrict__ addp,
                        const unsigned int* __restrict__ maxb,
                        const float* __restrict__ cnt,
                        const float* __restrict__ w1, const float* __restrict__ b1,
                        const float* __restrict__ w2, const float* __restrict__ b2,
                        float* __restrict__ out) {
  int g = threadIdx.x;
  if (g >= Gg) return;
  float gv[3 * Hd];
  float inv = 1.0f / fmaxf(cnt[g], 1.0f);
  for (int f = 0; f < Hd; ++f) {
    float s = addp[g * Hd + f];
    gv[f] = s;
    gv[Hd + f] = s * inv;
    gv[2 * Hd + f] = __uint_as_float(maxb[g * Hd + f]);
  }
  float hid[Hd];
  for (int j = 0; j < Hd; ++j) {
    float s = b1[j];
    for (int i = 0; i < 3 * Hd; ++i) s = fmaf(gv[i], w1[i * Hd + j], s);
    hid[j] = s > 0.0f ? s : 0.0f;
  }
  float l0 = b2[0], l1 = b2[1];
  for (int j = 0; j < Hd; ++j) {
    l0 = fmaf(hid[j], w2[j * 2 + 0], l0);
    l1 = fmaf(hid[j], w2[j * 2 + 1], l1);
  }
  float m = fmaxf(l0, l1);
  float lse = m + logf(expf(l0 - m) + expf(l1 - m));
  out[g * 2 + 0] = l0 - lse;
  out[g * 2 + 1] = l1 - lse;
}

// ---------------------------------------------------------------------------
extern "C" void kernel_launch(void* const* d_in, const int* in_sizes, int n_in,
                              void* d_out, int out_size, void* d_ws, size_t ws_size,
                              hipStream_t stream) {
  const float* x     = (const float*)d_in[0];
  const int*   ei    = (const int*)  d_in[1];
  const float* ea    = (const float*)d_in[2];
  const int*   batch = (const int*)  d_in[3];
  const float* nw1 = (const float*)d_in[4];  const float* nb1 = (const float*)d_in[5];
  const float* rt1 = (const float*)d_in[6];  const float* bi1 = (const float*)d_in[7];
  const float* nw2 = (const float*)d_in[8];  const float* nb2 = (const float*)d_in[9];
  const float* rt2 = (const float*)d_in[10]; const float* bi2 = (const float*)d_in[11];
  const float* nw3 = (const float*)d_in[12]; const float* nb3 = (const float*)d_in[13];
  const float* rt3 = (const float*)d_in[14]; const float* bi3 = (const float*)d_in[15];
  const float* l1w = (const float*)d_in[16]; const float* l1b = (const float*)d_in[17];
  const float* l2w = (const float*)d_in[18]; const float* l2b = (const float*)d_in[19];
  float* out = (float*)d_out;

  float* ws   = (float*)d_ws;
  float* P    = ws;                 // Nn*32
  float* Q    = P    + Nn * Hd;     // Nn*32
  float* R    = Q    + Nn * Hd;     // Nn*32
  float* agg  = R    + Nn * Hd;     // Nn*32
  float* hbuf = agg  + Nn * Hd;     // Nn*32
  float* addp = hbuf + Nn * Hd;     // Gg*32
  unsigned int* maxb = (unsigned int*)(addp + Gg * Hd);  // Gg*32
  float* cnt  = (float*)(maxb + Gg * Hd);                // Gg

  const int gemm_blocks = (Nn / 16 + 3) / 4;       // 157
  const int nh          = Nn * Hd;                 // 320000
  const int nh_blocks   = (nh + 255) / 256;
  const int e_blocks    = (Ne * 8 + 255) / 256;
  const int pool_elems  = Gg * Hd * 2 + Gg;        // addp + maxb + cnt (contiguous)

  // ---- layer 1 (K = 33) ----
  gemm3_wmma<33><<<gemm_blocks, 128, 0, stream>>>(x, nw1, nb1, rt1, P, Q, R);
  fill_zero<<<nh_blocks, 256, 0, stream>>>(agg, nh);
  edge_scatter<<<e_blocks, 256, 0, stream>>>(ei, ea, P, Q, agg);
  node_update<<<nh_blocks, 256, 0, stream>>>(agg, R, bi1, hbuf);

  // ---- layer 2 (K = 32) ----
  gemm3_wmma<32><<<gemm_blocks, 128, 0, stream>>>(hbuf, nw2, nb2, rt2, P, Q, R);
  fill_zero<<<nh_blocks, 256, 0, stream>>>(agg, nh);
  edge_scatter<<<e_blocks, 256, 0, stream>>>(ei, ea, P, Q, agg);
  node_update<<<nh_blocks, 256, 0, stream>>>(agg, R, bi2, hbuf);

  // ---- layer 3 (K = 32) ----
  gemm3_wmma<32><<<gemm_blocks, 128, 0, stream>>>(hbuf, nw3, nb3, rt3, P, Q, R);
  fill_zero<<<nh_blocks, 256, 0, stream>>>(agg, nh);
  edge_scatter<<<e_blocks, 256, 0, stream>>>(ei, ea, P, Q, agg);
  node_update<<<nh_blocks, 256, 0, stream>>>(agg, R, bi3, hbuf);

  // ---- pooling + readout ----
  fill_zero<<<(pool_elems + 255) / 256, 256, 0, stream>>>(addp, pool_elems);
  pool_kernel<<<nh_blocks, 256, 0, stream>>>(hbuf, batch, addp, maxb, cnt);
  readout<<<1, 64, 0, stream>>>(addp, maxb, cnt, l1w, l1b, l2w, l2b, out);
}